// loss_layer_PDE_60851096649865
// MI455X (gfx1250) — compile-verified
//
#include <hip/hip_runtime.h>

typedef __attribute__((ext_vector_type(2))) float v2f;
typedef __attribute__((ext_vector_type(8))) float v8f;

#define RHS_COEF (-39.478417604357434f)   /* -4*pi^2 */
#define IPT 4                             /* samples per thread */

__device__ __forceinline__ float fast_tanh(float z) {
#if __has_builtin(__builtin_amdgcn_tanhf)
    return __builtin_amdgcn_tanhf(z);          // v_tanh_f32 (gfx1250)
#else
    // tanh(z) = 1 - 2/(exp2(2z*log2e)+1); saturates correctly at +-1
    float e = __builtin_amdgcn_exp2f(z * 2.8853900817779268f);
    return 1.0f - 2.0f * __builtin_amdgcn_rcpf(e + 1.0f);
#endif
}

// ---------------------------------------------------------------------------
// Main kernel: fused h/ddu/residual^2 + per-block reduction.
// Each block covers 256*IPT = 1024 samples; partial sum -> ws[blockIdx.x].
// Loads are clamped-in-bounds (no exec-mask branches); invalid lanes are
// masked out of the sum with a cndmask instead.
// ---------------------------------------------------------------------------
__global__ __launch_bounds__(256) void pinn_pde_main(
    const float* __restrict__ x,  const float* __restrict__ W1,
    const float* __restrict__ b1, const float* __restrict__ W2,
    const float* __restrict__ kptr, float* __restrict__ partial, int n)
{
    __shared__ float red[8];
    const int tid  = threadIdx.x;
    const int base = blockIdx.x * (256 * IPT);

    float xi[IPT], acc[IPT];
    bool  valid[IPT];
#pragma unroll
    for (int i = 0; i < IPT; ++i) {
        int idx  = base + i * 256 + tid;          // coalesced
        valid[i] = (idx < n);
        idx      = valid[i] ? idx : (n - 1);      // always in-bounds
        xi[i]    = x[idx];                        // straight-line load
        acc[i]   = 0.0f;
    }
    const float kk = kptr[0];

    // j-loop outermost: one (w,b,e) triple serves IPT samples.
#pragma unroll
    for (int j = 0; j < 40; ++j) {
        const float w = W1[j];                    // uniform -> s_load
        const float b = b1[j];
        const float e = -2.0f * w * w * W2[j];
#pragma unroll
        for (int i = 0; i < IPT; ++i) {
            float h = fast_tanh(fmaf(xi[i], w, b));
            float t = h * h;
            acc[i]  = fmaf(e * h, 1.0f - t, acc[i]);   // += e*h*(1-h^2)
        }
    }

    float sum = 0.0f;
#pragma unroll
    for (int i = 0; i < IPT; ++i) {
        // v_sin_f32 computes sin(2*pi*src): exactly sin(2*pi*x), no reduction
        float rhs = RHS_COEF * __builtin_amdgcn_sinf(xi[i]);
        float r   = fmaf(kk, acc[i], -rhs);            // k*ddu - rhs
        sum += valid[i] ? r * r : 0.0f;                // v_cndmask, no branch
    }

    // wave32 tree reduce, then 8 wave leaders through LDS
#pragma unroll
    for (int off = 16; off > 0; off >>= 1) sum += __shfl_xor(sum, off, 32);
    if ((tid & 31) == 0) red[tid >> 5] = sum;
    __syncthreads();
    if (tid == 0) {
        float s = 0.0f;
#pragma unroll
        for (int w = 0; w < 8; ++w) s += red[w];
        partial[blockIdx.x] = s;
    }
}

// ---------------------------------------------------------------------------
// Final kernel: one wave. WMMA-chained reduction of the (zero-padded) block
// partials (A = ones 16x4, B = 64 partials/chunk, C accumulates), then the
// boundary loss. Deterministic: fixed-order WMMA chain + shfl tree.
// nPadded is a multiple of 64 and the pad is pre-zeroed, so loads are
// unconditional.
// ---------------------------------------------------------------------------
__global__ __launch_bounds__(32) void pinn_pde_final(
    const float* __restrict__ partial, int nPadded, int n,
    const float* __restrict__ x0, const float* __restrict__ W1,
    const float* __restrict__ b1, const float* __restrict__ W2,
    const float* __restrict__ b2, float* __restrict__ out)
{
    const int lane = threadIdx.x;

    v2f A; A.x = 1.0f; A.y = 1.0f;           // 16x4 all-ones
    v8f C = {};
    const int nChunks = nPadded >> 6;
    for (int c = 0; c < nChunks; ++c) {
        int i0 = (c << 6) + lane;
        v2f B;
        B.x = partial[i0];
        B.y = partial[i0 + 32];
        // D[m,n] = C[m,n] + sum_k B[k,n]  (A==1); total over D row 0 == sum(B)
        C = __builtin_amdgcn_wmma_f32_16x16x4_f32(
                /*neg_a=*/false, A, /*neg_b=*/false, B,
                /*c_mod=*/(short)0, C, /*reuse_a=*/false, /*reuse_b=*/false);
    }

    // D row 0 (M=0) lives in C[0] on lanes 0..15; sum those 16 lanes.
    float v = C[0];
    v += __shfl_xor(v, 1, 32);
    v += __shfl_xor(v, 2, 32);
    v += __shfl_xor(v, 4, 32);
    v += __shfl_xor(v, 8, 32);
    float pde = v / (float)n;                // lane 0 holds mean residual^2

    // Boundary loss: 2 points x 40 units, computed redundantly (uniform).
    float bc = 0.0f;
#pragma unroll
    for (int p = 0; p < 2; ++p) {
        float xp = x0[p];
        float u  = b2[0];
#pragma unroll
        for (int j = 0; j < 40; ++j)
            u += fast_tanh(fmaf(xp, W1[j], b1[j])) * W2[j];
        bc += u * u;                          // W_BC = 1.0
    }

    if (lane == 0) out[0] = bc + pde;
}

extern "C" void kernel_launch(void* const* d_in, const int* in_sizes, int n_in,
                              void* d_out, int out_size, void* d_ws, size_t ws_size,
                              hipStream_t stream)
{
    const float* x  = (const float*)d_in[0];
    const float* x0 = (const float*)d_in[1];
    const float* W1 = (const float*)d_in[2];
    const float* b1 = (const float*)d_in[3];
    const float* W2 = (const float*)d_in[4];
    const float* b2 = (const float*)d_in[5];
    const float* k  = (const float*)d_in[6];
    float* out = (float*)d_out;

    const int n = in_sizes[0];
    const int blocks  = (n + 256 * IPT - 1) / (256 * IPT);  // ~977 for n=1e6
    const int nPadded = (blocks + 63) & ~63;                // multiple of 64
    float* partial = (float*)d_ws;                          // nPadded*4 bytes

    // Zero the pad region so the WMMA reduction can load unconditionally.
    // (hipMemsetAsync on `stream` is graph-capture legal.)
    if (nPadded > blocks)
        hipMemsetAsync(partial + blocks, 0, (size_t)(nPadded - blocks) * sizeof(float), stream);

    pinn_pde_main<<<blocks, 256, 0, stream>>>(x, W1, b1, W2, k, partial, n);
    pinn_pde_final<<<1, 32, 0, stream>>>(partial, nPadded, n, x0, W1, b1, W2, b2, out);
}